// GCN_21414706938556
// MI455X (gfx1250) — compile-verified
//
#include <hip/hip_runtime.h>

// ---------------------------------------------------------------------------
// GCN (2x GraphConv, norm='both') for MI455X / gfx1250 (wave32, WMMA).
//
// Pipeline (all on `stream`, single-stream ordering gives dependencies):
//   0) zero  : deg_out, deg_in, agg1, agg2   (contiguous ws prefix)
//   1) degree: u32 atomic histograms over src / dst
//   2) norm  : norm = rsqrt(max(deg,1))
//   3) scatter1: agg1[dst] += x[src]*norm_src      (E x 64  f32 atomics)
//   4) gemm1  : h1 = relu((agg1*norm_dst) @ W1 + b1)   [fp32 WMMA 16x16x4]
//   5) scatter2: agg2[dst] += h1[src]*norm_src     (E x 128 f32 atomics)
//   6) gemm2  : out = (agg2*norm_dst) @ W2 + b2        [fp32 WMMA 16x16x4]
// ---------------------------------------------------------------------------

typedef __attribute__((ext_vector_type(2))) float v2f;
typedef __attribute__((ext_vector_type(8))) float v8f;

__global__ void zero_words_kernel(unsigned* __restrict__ p, long long n) {
  long long i = (long long)blockIdx.x * blockDim.x + threadIdx.x;
  long long stride = (long long)gridDim.x * blockDim.x;
  for (; i < n; i += stride) p[i] = 0u;
}

__global__ void degree_kernel(const int* __restrict__ src, const int* __restrict__ dst,
                              unsigned* __restrict__ dout, unsigned* __restrict__ din,
                              int E) {
  int t = blockIdx.x * blockDim.x + threadIdx.x;
  if (t < E) {
    atomicAdd(&dout[src[t]], 1u);
    atomicAdd(&din[dst[t]], 1u);
  }
}

__global__ void norm_kernel(const unsigned* __restrict__ dout, const unsigned* __restrict__ din,
                            float* __restrict__ ns, float* __restrict__ nd, int N) {
  int i = blockIdx.x * blockDim.x + threadIdx.x;
  if (i < N) {
    ns[i] = rsqrtf(fmaxf((float)dout[i], 1.0f));
    nd[i] = rsqrtf(fmaxf((float)din[i], 1.0f));
  }
}

// One thread per (edge, feature). Consecutive D threads share an edge, so the
// feature gather and the atomic adds are cacheline-coalesced; src/dst loads
// are wave-uniform-ish broadcasts. norm_src is folded into the gather.
__global__ void scatter_kernel(const float* __restrict__ feat, const float* __restrict__ ns,
                               const int* __restrict__ src, const int* __restrict__ dst,
                               float* __restrict__ agg, int E, int logD) {
  const int D = 1 << logD;
  long long total = (long long)E << logD;
  long long t = (long long)blockIdx.x * blockDim.x + threadIdx.x;
  if (t < total) {
    int e = (int)(t >> logD);
    int f = (int)t & (D - 1);
    int s = src[e];
    int d = dst[e];
    float v = feat[(size_t)s * D + f] * ns[s];
    atomicAdd(&agg[(size_t)d * D + f], v);
  }
}

// out[M x Ncols] = (A[M x K] * nd[M]) @ W[K x Ncols] + bias (+ReLU).
// Block = 128 threads = 4 waves; each wave owns a 16x16 output tile
// (block covers 16 rows x 64 cols). Requires M % 16 == 0 handled softly via
// row clamp on loads + guarded stores; Ncols % 64 == 0 (128 / 64 here).
//
// fp32 WMMA fragment layout (ISA 7.12.2):
//   A 16x4 : lanes 0-15 -> M=lane,  v0=K+0, v1=K+1 ; lanes 16-31 -> K+2,K+3
//   B 4x16 : lanes 0-15 -> N=lane,  v0=K+0, v1=K+1 ; lanes 16-31 -> K+2,K+3
//   C/D    : VGPR r, lanes 0-15 -> (M=r, N=lane); lanes 16-31 -> (M=r+8)
template <bool RELU>
__global__ void gemm_wmma_kernel(const float* __restrict__ A, const float* __restrict__ nd,
                                 const float* __restrict__ W, const float* __restrict__ bias,
                                 float* __restrict__ out, int M, int K, int Ncols) {
  const int lane = threadIdx.x & 31;
  const int wave = threadIdx.x >> 5;
  const int col0 = blockIdx.x * 64 + wave * 16;
  const int row0 = blockIdx.y * 16;
  if (col0 >= Ncols) return;  // wave-uniform

  const int half = lane >> 4;   // which K pair this lane supplies
  const int l16  = lane & 15;

  int arow = row0 + l16;
  if (arow >= M) arow = M - 1;          // clamp keeps EXEC all-ones for WMMA
  const float nscale = nd[arow];
  const float* __restrict__ Arow = A + (size_t)arow * K;
  const int bcol = col0 + l16;

  v8f c = {};
  for (int k = 0; k < K; k += 4) {
    const int ka = k + half * 2;
    v2f a, b;
    a.x = Arow[ka]     * nscale;
    a.y = Arow[ka + 1] * nscale;
    b.x = W[(size_t)ka       * Ncols + bcol];
    b.y = W[(size_t)(ka + 1) * Ncols + bcol];
    // 8 args: (neg_a, A, neg_b, B, c_mod, C, reuse_a, reuse_b)
    c = __builtin_amdgcn_wmma_f32_16x16x4_f32(false, a, false, b,
                                              (short)0, c, false, false);
  }

  const float bv = bias[bcol];
#pragma unroll
  for (int r = 0; r < 8; ++r) {
    const int row = row0 + half * 8 + r;
    if (row < M) {
      float v = c[r] + bv;
      if (RELU) v = fmaxf(v, 0.0f);
      out[(size_t)row * Ncols + bcol] = v;
    }
  }
}

extern "C" void kernel_launch(void* const* d_in, const int* in_sizes, int n_in,
                              void* d_out, int out_size, void* d_ws, size_t ws_size,
                              hipStream_t stream) {
  const float* in_feat = (const float*)d_in[0];  // [N, IN]
  const float* W1      = (const float*)d_in[1];  // [IN, HID]
  const float* b1      = (const float*)d_in[2];  // [HID]
  const float* W2      = (const float*)d_in[3];  // [HID, IN]
  const float* b2      = (const float*)d_in[4];  // [IN]
  const int*   src     = (const int*)d_in[5];    // [E]
  const int*   dst     = (const int*)d_in[6];    // [E]
  // d_in[7] = n_nodes scalar lives on device; derive dims on host instead.

  const int HID = in_sizes[2];            // 128
  const int IN  = in_sizes[4];            // 64
  const int N   = in_sizes[0] / IN;       // 50000
  const int E   = in_sizes[5];            // 800000
  int logIN = 0;  while ((1 << logIN) < IN)  ++logIN;
  int logHID = 0; while ((1 << logHID) < HID) ++logHID;

  // Workspace layout: buffers needing zero-init first (contiguous prefix).
  char* ws = (char*)d_ws;
  size_t off = 0;
  unsigned* deg_out = (unsigned*)(ws + off); off += (size_t)N * 4;
  unsigned* deg_in  = (unsigned*)(ws + off); off += (size_t)N * 4;
  float* agg1 = (float*)(ws + off); off += (size_t)N * IN * 4;
  float* agg2 = (float*)(ws + off); off += (size_t)N * HID * 4;
  float* norm_src = (float*)(ws + off); off += (size_t)N * 4;
  float* norm_dst = (float*)(ws + off); off += (size_t)N * 4;
  float* h1 = (float*)(ws + off); off += (size_t)N * HID * 4;

  // 0) zero the prefix (deg_out, deg_in, agg1, agg2)
  long long zwords = (long long)N * (2 + IN + HID);
  int zb = (int)((zwords + 255) / 256);
  zero_words_kernel<<<zb, 256, 0, stream>>>((unsigned*)ws, zwords);

  // 1) degrees
  degree_kernel<<<(E + 255) / 256, 256, 0, stream>>>(src, dst, deg_out, deg_in, E);

  // 2) norms
  norm_kernel<<<(N + 255) / 256, 256, 0, stream>>>(deg_out, deg_in, norm_src, norm_dst, N);

  // 3) layer-1 scatter: agg1[dst] += x[src] * norm_src[src]
  {
    long long total = (long long)E << logIN;
    scatter_kernel<<<(int)((total + 255) / 256), 256, 0, stream>>>(
        in_feat, norm_src, src, dst, agg1, E, logIN);
  }

  // 4) layer-1 GEMM + bias + relu  (h1 = relu((agg1*nd) @ W1 + b1))
  {
    dim3 grid(HID / 64, (N + 15) / 16);
    gemm_wmma_kernel<true><<<grid, 128, 0, stream>>>(agg1, norm_dst, W1, b1, h1,
                                                     N, IN, HID);
  }

  // 5) layer-2 scatter: agg2[dst] += h1[src] * norm_src[src]
  {
    long long total = (long long)E << logHID;
    scatter_kernel<<<(int)((total + 255) / 256), 256, 0, stream>>>(
        h1, norm_src, src, dst, agg2, E, logHID);
  }

  // 6) layer-2 GEMM + bias  (out = (agg2*nd) @ W2 + b2)
  {
    dim3 grid(IN / 64, (N + 15) / 16);
    gemm_wmma_kernel<false><<<grid, 128, 0, stream>>>(agg2, norm_dst, W2, b2,
                                                      (float*)d_out, N, HID, IN);
  }
}